// ProductLayer_46059229282761
// MI455X (gfx1250) — compile-verified
//
#include <hip/hip_runtime.h>

typedef __attribute__((ext_vector_type(2))) float v2f;
typedef __attribute__((ext_vector_type(8))) float v8f;

// One wave handles 16 consecutive output segments.
// C (16x16 f32) += A (16x4 selection 0/1) x B (4 gathered rows x 16 cols)
// via V_WMMA_F32_16X16X4_F32; 2x unrolled ping-pong pipeline so each WMMA's
// B operand is a full unrolled-body old -> partial s_wait_loadcnt, loads overlap.
__global__ __launch_bounds__(256) void seg_gather_sum_wmma(
    const float* __restrict__ x,     // [n_nodes, 16]
    const int*   __restrict__ ptrs,  // [nnz]
    const int*   __restrict__ csr,   // [n_seg + 1]
    float*       __restrict__ out,   // [n_seg, 16]
    int n_seg, int nnz)
{
    const int lane  = threadIdx.x & 31;
    const int gwave = blockIdx.x * (blockDim.x >> 5) + (threadIdx.x >> 5);
    const int s0    = gwave * 16;              // first segment of this wave's window
    if (s0 >= n_seg) return;                   // wave-uniform exit: EXEC all-1s for WMMA

    const int ml = lane & 15;                  // M (A-row) / N (B,D column) index
    const int kh = lane >> 4;                  // K half: 0 -> K{0,1}, 1 -> K{2,3}

    const int hi   = (s0 + 16 <= n_seg) ? 16 : (n_seg - s0);
    const int base = csr[s0];
    const int end  = csr[s0 + hi];
    // lane l holds window boundary csr[s0+1+l] (replicated across halves, clamped)
    const int bidx     = (1 + ml <= hi) ? (1 + ml) : hi;
    const int boundary = csr[s0 + bidx];

    v8f c = {};                                // 16x16 f32 accumulator

    if (base < end) {
        const int last = end - 1;              // clamp target: always a valid row
        const int koff = 2 * kh;               // this lane's K offset within a chunk

        // ---- pipeline prologue ----
        // indices of chunk 0 and chunk 1 (clamped -> unconditional loads)
        int i0 = base + koff,      i1 = i0 + 1;
        int i2 = base + 4 + koff,  i3 = i2 + 1;
        int pa0 = ptrs[i0 < end ? i0 : last];
        int pb0 = ptrs[i1 < end ? i1 : last];
        int pa1 = ptrs[i2 < end ? i2 : last];
        int pb1 = ptrs[i3 < end ? i3 : last];
        // rows of chunk 0 / chunk 1
        float b0x = x[pa0 * 16 + ml];          // 16 lanes -> 64B coalesced per row
        float b0y = x[pb0 * 16 + ml];
        float b1x = x[pa1 * 16 + ml];
        float b1y = x[pb1 * 16 + ml];
        // indices of chunk 2 / chunk 3
        int i4 = base + 8 + koff,  i5 = i4 + 1;
        int i6 = base + 12 + koff, i7 = i6 + 1;
        int pa2 = ptrs[i4 < end ? i4 : last];
        int pb2 = ptrs[i5 < end ? i5 : last];
        int pa3 = ptrs[i6 < end ? i6 : last];
        int pb3 = ptrs[i7 < end ? i7 : last];

        // invariants at top of body for j:
        //   (b0x,b0y)=rows(chunk j)   (b1x,b1y)=rows(chunk j+4)
        //   (pa2,pb2)=idx(chunk j+8)  (pa3,pb3)=idx(chunk j+12)
        for (int j = base; j < end; j += 8) {
            // prefetch the sequential index stream far ahead (global_prefetch_b8)
            int pf = j + 512;
            if (pf > nnz - 1) pf = nnz - 1;
            __builtin_prefetch(&ptrs[pf], 0, 1);

            // ---- half A: chunk j ----
            {
                const int cj0 = j + koff, cj1 = cj0 + 1;
                const unsigned bal0 = __builtin_amdgcn_ballot_w32(boundary <= cj0);
                const unsigned bal1 = __builtin_amdgcn_ballot_w32(boundary <= cj1);
                const int seg0 = __builtin_popcount(kh ? (bal0 >> 16) : (bal0 & 0xffffu));
                const int seg1 = __builtin_popcount(kh ? (bal1 >> 16) : (bal1 & 0xffffu));
                v2f a, b;
                a.x = (cj0 < end && seg0 == ml) ? 1.0f : 0.0f;
                a.y = (cj1 < end && seg1 == ml) ? 1.0f : 0.0f;
                b.x = b0x; b.y = b0y;
                c = __builtin_amdgcn_wmma_f32_16x16x4_f32(
                        false, a, false, b, (short)0, c, false, false);
            }
            // b0 consumed -> reload same registers with rows(chunk j+8)
            b0x = x[pa2 * 16 + ml];
            b0y = x[pb2 * 16 + ml];
            // refresh idx(chunk j+16)
            {
                int f0 = j + 16 + koff, f1 = f0 + 1;
                pa2 = ptrs[f0 < end ? f0 : last];
                pb2 = ptrs[f1 < end ? f1 : last];
            }

            // ---- half B: chunk j+4 (may be all-tail: A=0, harmless) ----
            {
                const int cj0 = j + 4 + koff, cj1 = cj0 + 1;
                const unsigned bal0 = __builtin_amdgcn_ballot_w32(boundary <= cj0);
                const unsigned bal1 = __builtin_amdgcn_ballot_w32(boundary <= cj1);
                const int seg0 = __builtin_popcount(kh ? (bal0 >> 16) : (bal0 & 0xffffu));
                const int seg1 = __builtin_popcount(kh ? (bal1 >> 16) : (bal1 & 0xffffu));
                v2f a, b;
                a.x = (cj0 < end && seg0 == ml) ? 1.0f : 0.0f;
                a.y = (cj1 < end && seg1 == ml) ? 1.0f : 0.0f;
                b.x = b1x; b.y = b1y;
                c = __builtin_amdgcn_wmma_f32_16x16x4_f32(
                        false, a, false, b, (short)0, c, false, false);
            }
            // b1 consumed -> reload with rows(chunk j+12)
            b1x = x[pa3 * 16 + ml];
            b1y = x[pb3 * 16 + ml];
            // refresh idx(chunk j+20)
            {
                int f0 = j + 20 + koff, f1 = f0 + 1;
                pa3 = ptrs[f0 < end ? f0 : last];
                pb3 = ptrs[f1 < end ? f1 : last];
            }
        }
    }

    // D layout: VGPR r -> segment s0 + r + 8*kh, column ml; 64B coalesced per row
    const int row0 = s0 + 8 * kh;
    if (hi == 16) {
        // uniform fast path: no per-row guards
#pragma unroll
        for (int r = 0; r < 8; ++r)
            out[(row0 + r) * 16 + ml] = c[r];
    } else {
#pragma unroll
        for (int r = 0; r < 8; ++r) {
            const int row = row0 + r;
            if (row < n_seg) out[row * 16 + ml] = c[r];
        }
    }
}

extern "C" void kernel_launch(void* const* d_in, const int* in_sizes, int n_in,
                              void* d_out, int out_size, void* d_ws, size_t ws_size,
                              hipStream_t stream) {
    const float* x    = (const float*)d_in[0];
    const int*   ptrs = (const int*)d_in[1];
    const int*   csr  = (const int*)d_in[2];
    float*       out  = (float*)d_out;

    const int nnz   = in_sizes[1];
    const int n_seg = in_sizes[2] - 1;

    const int waves_needed  = (n_seg + 15) / 16;      // 16 segments per wave
    const int waves_per_blk = 8;                      // 256 threads = 8 wave32
    const int blocks = (waves_needed + waves_per_blk - 1) / waves_per_blk;

    seg_gather_sum_wmma<<<blocks, 256, 0, stream>>>(x, ptrs, csr, out, n_seg, nnz);
}